// NUFFTOperator_566935683359
// MI455X (gfx1250) — compile-verified
//
#include <hip/hip_runtime.h>
#include <math.h>

// ---------------------------------------------------------------------------
// NUFFT type-2 for MI455X (gfx1250, wave32).
//   K1: deapodize + zero-pad + fftshift-roll  -> buf0 (float2 grids)
//   K2: 512-pt FFT along rows via four-step (WMMA DFT-16 + LDS Stockham-32),
//       transposed store.  Run twice -> full FFT2 in natural layout.
//       Row staging uses GLOBAL_LOAD_ASYNC_TO_LDS_B64 when available.
//   K3: Kaiser-Bessel weights + wrapped indices per k-point (Bessel I0).
//   K4: 6x6 weighted gather per (channel, point) -> out (B,C,M,2) float32.
// Workspace requirement: 2*16*512*512*8  +  4*65536*6*4  ~= 73.5 MB.
// ---------------------------------------------------------------------------

typedef __attribute__((ext_vector_type(2))) float v2f;
typedef __attribute__((ext_vector_type(8))) float v8f;
typedef __attribute__((ext_vector_type(2))) int   v2i;

#define N_IMG   16          // B*C
#define GSZ     512
#define GG      (GSZ*GSZ)   // 262144
#define M_PTS   65536
#define ALPHA_F 14.04f      // 2.34 * W, W = 6
#define PI_F    3.14159265358979f

#if __has_builtin(__builtin_amdgcn_global_load_async_to_lds_b64) && \
    __has_builtin(__builtin_amdgcn_s_wait_asynccnt)
#define HAVE_ASYNC_LDS 1
typedef __attribute__((address_space(1))) v2i gls_v2i;   // global b64 payload
typedef __attribute__((address_space(3))) v2i lds_v2i;   // LDS b64 payload
#else
#define HAVE_ASYNC_LDS 0
#endif

// ---- Kaiser-Bessel helpers -------------------------------------------------

__device__ __forceinline__ float bessel_i0f(float x) {
  float ax = fabsf(x);
  if (ax < 3.75f) {
    float t = ax * (1.0f / 3.75f);
    t *= t;
    return 1.0f + t * (3.5156229f + t * (3.0899424f + t * (1.2067492f +
                 t * (0.2659732f + t * (0.0360768f + t * 0.0045813f)))));
  }
  float t = 3.75f / ax;
  return (__expf(ax) * __frsqrt_rn(ax)) *
         (0.39894228f + t * (0.01328592f + t * (0.00225319f + t * (-0.00157565f +
          t * (0.00916281f + t * (-0.02057706f + t * (0.02635537f +
          t * (-0.01647633f + t * 0.00392377f))))))));
}

__device__ __forceinline__ float kb_ft(float t) {
  float pwt = PI_F * 6.0f * t;
  float s2  = ALPHA_F * ALPHA_F - pwt * pwt;
  float z   = sqrtf(fabsf(s2)) + 1e-12f;
  if (s2 >= 0.0f) return (__expf(z) - __expf(-z)) * 0.5f / z; // sinh(z)/z
  return __sinf(z) / z;
}

// ---- K1: deapodize + pad + roll -------------------------------------------

__global__ __launch_bounds__(256) void deapod_pad_roll(
    const float* __restrict__ x, float2* __restrict__ g) {
  int gid = blockIdx.x * 256 + threadIdx.x;     // 16 * 512 * 512 threads
  int img = gid >> 18;
  int rem = gid & (GG - 1);
  int gy  = rem >> 9;
  int gx  = rem & 511;
  int sy  = (gy + 128) & 511;                   // inverse of roll(-128)
  int sx  = (gx + 128) & 511;
  float v = 0.0f;
  if (sy < 256 && sx < 256) {
    float ty = (float)(sy - 128) * (1.0f / 512.0f);
    float tx = (float)(sx - 128) * (1.0f / 512.0f);
    v = x[img * 65536 + sy * 256 + sx] / (kb_ft(ty) * kb_ft(tx));
  }
  g[gid] = make_float2(v, 0.0f);
}

// ---- K2: FFT-512 along rows, transposed store -----------------------------
// 512 = 16 x 32 four-step:  Y[k1,b]   = sum_a x[32a+b] W16^(a*k1)   (WMMA)
//                           Z[k1,b]   = Y[k1,b] * W512^(k1*b)
//                           X[k1+16c] = sum_b Z[k1,b] W32^(b*c)     (Stockham)
// One wave (32 lanes) per row; 4 rows (waves) per block.

__global__ __launch_bounds__(128) void fft512_rows_wmma(
    const float2* __restrict__ in, float2* __restrict__ out) {
  __shared__ float2 aC[4][512];
  __shared__ float2 bC[4][512];

  const int tid  = threadIdx.x;
  const int w    = tid >> 5;          // wave within block
  const int lane = tid & 31;
  const int img  = blockIdx.y;
  const int r0   = blockIdx.x << 2;   // first row of this block
  const int row  = r0 + w;

  const float2* src = in + (size_t)img * GG + (size_t)row * GSZ;

  // Stage row into LDS.  Prefer the CDNA5 async global->LDS path (ASYNCcnt).
#if HAVE_ASYNC_LDS
#pragma unroll
  for (int j = 0; j < 16; ++j) {
    int n = lane + 32 * j;
    __builtin_amdgcn_global_load_async_to_lds_b64(
        (gls_v2i*)(src + n), (lds_v2i*)&aC[w][n], 0, 0);
  }
  __builtin_amdgcn_s_wait_asynccnt(0);
#else
#pragma unroll
  for (int j = 0; j < 16; ++j) {
    int n = lane + 32 * j;
    aC[w][n] = src[n];
  }
#endif
  __syncthreads();

  const int hi  = lane >> 4;          // which half-wave
  const int l15 = lane & 15;

  // --- DFT-16 over 'a' for all 32 columns 'b' via WMMA f32 16x16x4 ---------
  // A = DFT16 matrix (computed in-lane), B = data tile X[a, b-tile].
  v8f Dr0 = {}, Di0 = {}, Dr1 = {}, Di1 = {};
#pragma unroll
  for (int s = 0; s < 4; ++s) {
    const int a0 = 4 * s + 2 * hi;            // K indices this lane supplies
    const float k1f = (float)l15;             // A-operand row M = k1
    float ph0 = (2.0f * PI_F / 16.0f) * k1f * (float)a0;
    float dph = (2.0f * PI_F / 16.0f) * k1f;
    float c0, s0, c1, s1;
    __sincosf(ph0, &s0, &c0);
    __sincosf(ph0 + dph, &s1, &c1);
    v2f Fc  = { c0,  c1 };                    // Re F = cos
    v2f Fs  = { s0,  s1 };                    // -Im F = sin
    v2f Fns = { -s0, -s1 };

    // B-operand: lane supplies X[K, bc] for K = a0, a0+1 (mirrored C/D layout)
    {
      int bc = l15;                           // tile 0: b in [0,16)
      float2 x0 = aC[w][32 * a0 + bc];
      float2 x1 = aC[w][32 * (a0 + 1) + bc];
      v2f Xr = { x0.x, x1.x };
      v2f Xi = { x0.y, x1.y };
      Dr0 = __builtin_amdgcn_wmma_f32_16x16x4_f32(false, Fc,  false, Xr, (short)0, Dr0, false, false);
      Dr0 = __builtin_amdgcn_wmma_f32_16x16x4_f32(false, Fs,  false, Xi, (short)0, Dr0, false, false);
      Di0 = __builtin_amdgcn_wmma_f32_16x16x4_f32(false, Fc,  false, Xi, (short)0, Di0, false, false);
      Di0 = __builtin_amdgcn_wmma_f32_16x16x4_f32(false, Fns, false, Xr, (short)0, Di0, false, false);
    }
    {
      int bc = 16 + l15;                      // tile 1: b in [16,32)
      float2 x0 = aC[w][32 * a0 + bc];
      float2 x1 = aC[w][32 * (a0 + 1) + bc];
      v2f Xr = { x0.x, x1.x };
      v2f Xi = { x0.y, x1.y };
      Dr1 = __builtin_amdgcn_wmma_f32_16x16x4_f32(false, Fc,  false, Xr, (short)0, Dr1, false, false);
      Dr1 = __builtin_amdgcn_wmma_f32_16x16x4_f32(false, Fs,  false, Xi, (short)0, Dr1, false, false);
      Di1 = __builtin_amdgcn_wmma_f32_16x16x4_f32(false, Fc,  false, Xi, (short)0, Di1, false, false);
      Di1 = __builtin_amdgcn_wmma_f32_16x16x4_f32(false, Fns, false, Xr, (short)0, Di1, false, false);
    }
  }

  // --- Twiddle W512^(k1*b) and store Z[k1][b] back to LDS ------------------
#pragma unroll
  for (int v = 0; v < 8; ++v) {
    int k1 = v + 8 * hi;                      // C/D layout: M = v (+8 hi half)
    {
      int b = l15;
      float sn, cs;
      __sincosf((2.0f * PI_F / 512.0f) * (float)(k1 * b), &sn, &cs);
      float yr = Dr0[v], yi = Di0[v];
      aC[w][k1 * 32 + b] = make_float2(yr * cs + yi * sn, yi * cs - yr * sn);
    }
    {
      int b = 16 + l15;
      float sn, cs;
      __sincosf((2.0f * PI_F / 512.0f) * (float)(k1 * b), &sn, &cs);
      float yr = Dr1[v], yi = Di1[v];
      aC[w][k1 * 32 + b] = make_float2(yr * cs + yi * sn, yi * cs - yr * sn);
    }
  }
  __syncthreads();

  // --- 32-point Stockham DFT over b for each of 16 rows k1 -----------------
  {
    float2* sC = &aC[w][0];
    float2* dC = &bC[w][0];
#pragma unroll
    for (int t = 0; t < 5; ++t) {
      const int st = 1 << t;
#pragma unroll
      for (int it = 0; it < 8; ++it) {
        int task = lane * 8 + it;             // 256 butterflies / stage
        int rr   = task >> 4;                 // k1 row
        int pj   = task & 15;
        int kk   = pj & (st - 1);
        int j    = pj >> t;
        int base = rr * 32;
        float2 av = sC[base + pj];            // kk + st*j        == pj
        float2 bv = sC[base + pj + 16];       // kk + st*(j+16>>t) == pj+16
        float sn, cs;
        __sincosf(-(2.0f * PI_F / 32.0f) * (float)(j << t), &sn, &cs);
        float vr = av.x - bv.x, vi = av.y - bv.y;
        dC[base + kk + st * 2 * j] = make_float2(av.x + bv.x, av.y + bv.y);
        dC[base + kk + st * (2 * j + 1)] =
            make_float2(vr * cs - vi * sn, vr * sn + vi * cs);
      }
      __syncthreads();
      float2* tp = sC; sC = dC; dC = tp;
    }
  }
  // Result sits in bC (5 stages: a->b,b->a,a->b,b->a,a->b).

  // --- Transposed store: out[k][row] = X_row[k], k = k1 + 16c --------------
  {
    int ws  = tid & 3;                        // row within block
    int klo = tid >> 2;                       // 0..31
#pragma unroll
    for (int j = 0; j < 16; ++j) {
      int k = klo + 32 * j;
      int k1 = k & 15, c = k >> 4;
      out[(size_t)img * GG + (size_t)k * GSZ + (r0 + ws)] = bC[ws][k1 * 32 + c];
    }
  }
}

// ---- K3: Kaiser-Bessel weights + wrapped grid indices ---------------------

__global__ __launch_bounds__(256) void kb_weights(
    const float* __restrict__ ktraj,
    float* __restrict__ wts0, float* __restrict__ wts1,
    int* __restrict__ idx0, int* __restrict__ idx1) {
  int m = blockIdx.x * 256 + threadIdx.x;
  for (int d = 0; d < 2; ++d) {
    float tm   = ktraj[d * M_PTS + m] * (512.0f / (2.0f * PI_F));
    float base = floorf(tm);
    float* wd  = d ? wts1 : wts0;
    int*   id  = d ? idx1 : idx0;
#pragma unroll
    for (int t = 0; t < 6; ++t) {
      float p  = base + (float)(t - 2);       // offs = -2..3
      float dd = tm - p;
      float u  = 1.0f - (dd * (1.0f / 3.0f)) * (dd * (1.0f / 3.0f));
      float wv = 0.0f;
      if (u > 0.0f)
        wv = bessel_i0f(ALPHA_F * sqrtf(fmaxf(u, 1e-12f))) * (1.0f / 6.0f);
      wd[m * 6 + t] = wv;
      id[m * 6 + t] = ((int)p) & 511;         // mod 512 (power of two)
    }
  }
}

// ---- K4: 6x6 weighted gather (L2-resident grid) ---------------------------

__global__ __launch_bounds__(256) void interp_gather(
    const float2* __restrict__ Xk,
    const float* __restrict__ w0, const float* __restrict__ w1,
    const int* __restrict__ i0v, const int* __restrict__ i1v,
    float2* __restrict__ out) {
  int gid = blockIdx.x * 256 + threadIdx.x;   // 16 * 65536 threads
  int bc  = gid >> 16;
  int m   = gid & (M_PTS - 1);
  const float2* img = Xk + (size_t)bc * GG;

  float wa[6], wb[6]; int ia[6], ib[6];
#pragma unroll
  for (int t = 0; t < 6; ++t) {
    wa[t] = w0[m * 6 + t];
    wb[t] = w1[m * 6 + t];
    ia[t] = i0v[m * 6 + t] * 512;
    ib[t] = i1v[m * 6 + t];
  }
  float accr = 0.0f, acci = 0.0f;
#pragma unroll
  for (int i = 0; i < 6; ++i) {
    const float2* rowp = img + ia[i];
    float wi = wa[i];
#pragma unroll
    for (int j = 0; j < 6; ++j) {
      float2 v = rowp[ib[j]];
      float ww = wi * wb[j];
      accr = fmaf(ww, v.x, accr);
      acci = fmaf(ww, v.y, acci);
    }
  }
  out[(size_t)bc * M_PTS + m] = make_float2(accr, acci);
}

// ---- launcher -------------------------------------------------------------

extern "C" void kernel_launch(void* const* d_in, const int* in_sizes, int n_in,
                              void* d_out, int out_size, void* d_ws, size_t ws_size,
                              hipStream_t stream) {
  const float* x  = (const float*)d_in[0];   // (2,8,256,256) f32
  const float* kt = (const float*)d_in[1];   // (2,65536) f32

  float2* buf0 = (float2*)d_ws;                          // 33.5 MB
  float2* buf1 = buf0 + (size_t)N_IMG * GG;              // 33.5 MB
  float*  wts0 = (float*)(buf1 + (size_t)N_IMG * GG);    // 1.5 MB
  float*  wts1 = wts0 + (size_t)M_PTS * 6;               // 1.5 MB
  int*    idx0 = (int*)(wts1 + (size_t)M_PTS * 6);       // 1.5 MB
  int*    idx1 = idx0 + (size_t)M_PTS * 6;               // 1.5 MB

  deapod_pad_roll<<<(N_IMG * GG) / 256, 256, 0, stream>>>(x, buf0);

  dim3 fgrid(GSZ / 4, N_IMG);
  fft512_rows_wmma<<<fgrid, 128, 0, stream>>>(buf0, buf1);  // rows, T
  fft512_rows_wmma<<<fgrid, 128, 0, stream>>>(buf1, buf0);  // cols, T -> Xk

  kb_weights<<<M_PTS / 256, 256, 0, stream>>>(kt, wts0, wts1, idx0, idx1);

  interp_gather<<<(N_IMG * M_PTS) / 256, 256, 0, stream>>>(
      buf0, wts0, wts1, idx0, idx1, (float2*)d_out);
}